// MetricSimulator_35201551958460
// MI455X (gfx1250) — compile-verified
//
#include <hip/hip_runtime.h>

typedef __attribute__((ext_vector_type(2))) float v2f;
typedef __attribute__((ext_vector_type(8))) float v8f;

#define T_DIM 16384
#define B_DIM 64
#define E_DIM 4096

// ---------------------------------------------------------------------------
// Phase 1: alpha[t], beta[t] = sum over b of params[indices[t,b], {0,1}]
// One wave32 per t. Lane l handles b=l and b=l+32, then butterfly reduce.
// ---------------------------------------------------------------------------
__global__ __launch_bounds__(256) void coeff_reduce_kernel(
    const long long* __restrict__ indices,   // (T, 64) int64
    const float* __restrict__ params,        // (N, 2) f32
    float* __restrict__ alpha,
    float* __restrict__ beta) {
  const int wave = threadIdx.x >> 5;
  const int lane = threadIdx.x & 31;
  const int t = blockIdx.x * 8 + wave;

  const long long* row = indices + (size_t)t * B_DIM;
  const long long i0 = row[lane];
  const long long i1 = row[lane + 32];

  const float2* pp = (const float2*)params;
  const float2 c0 = pp[i0];
  const float2 c1 = pp[i1];

  float a = c0.x + c1.x;
  float b = c0.y + c1.y;
#pragma unroll
  for (int off = 16; off > 0; off >>= 1) {
    a += __shfl_xor(a, off, 32);
    b += __shfl_xor(b, off, 32);
  }
  if (lane == 0) {
    alpha[t] = a;
    beta[t] = b;
  }
}

// ---------------------------------------------------------------------------
// Phase 2: inclusive affine scan over T=16384 elements, single workgroup.
// compose(earlier=(pa,pb), later=(xa,xb)) = (xa*pa, xa*pb + xb)
// Each of 1024 threads owns 16 consecutive elements; block scan via LDS.
// ---------------------------------------------------------------------------
__global__ __launch_bounds__(1024) void affine_scan_kernel(
    const float* __restrict__ alpha, const float* __restrict__ beta,
    float* __restrict__ Aout, float* __restrict__ Bout) {
  __shared__ float sA[1024];
  __shared__ float sB[1024];
  const int tid = threadIdx.x;
  const int base = tid * 16;

  // thread-local aggregate
  float pa = 1.0f, pb = 0.0f;
#pragma unroll
  for (int i = 0; i < 16; ++i) {
    const float xa = alpha[base + i];
    const float xb = beta[base + i];
    pb = xa * pb + xb;
    pa = xa * pa;
  }
  sA[tid] = pa;
  sB[tid] = pb;
  __syncthreads();

  // Hillis-Steele inclusive scan of 1024 aggregates
  for (int off = 1; off < 1024; off <<= 1) {
    float qa = 1.0f, qb = 0.0f;
    if (tid >= off) { qa = sA[tid - off]; qb = sB[tid - off]; }
    __syncthreads();
    if (tid >= off) {
      const float ca = sA[tid];
      const float cb = sB[tid];
      sA[tid] = ca * qa;
      sB[tid] = ca * qb + cb;
    }
    __syncthreads();
  }

  // exclusive prefix for this thread, then re-walk and emit
  float ea = 1.0f, eb = 0.0f;
  if (tid > 0) { ea = sA[tid - 1]; eb = sB[tid - 1]; }
#pragma unroll
  for (int i = 0; i < 16; ++i) {
    const float xa = alpha[base + i];
    const float xb = beta[base + i];
    eb = xa * eb + xb;
    ea = xa * ea;
    Aout[base + i] = ea;
    Bout[base + i] = eb;
  }
}

// ---------------------------------------------------------------------------
// Phase 3: out[t,e] = A[t]*M_prev[e] + Bc[t] via V_WMMA_F32_16X16X4_F32.
// A-frag (16x4 f32): lanes 0-15 hold M=0..15, reg0=K0 col = A[t],
//   reg1=K1 col = Bc[t]; lanes 16-31 (K=2,3) zeroed.
// B-frag (4x16 f32): lanes 0-15 hold N=0..15, reg0=K0 row = M_prev[e],
//   reg1=K1 row = 1.0. Lanes 16-31 (K=2,3 rows) multiply A's zero columns,
//   so they may hold arbitrary *finite* values -> no masking needed at all.
// D = A*B = A[t]*M_prev[e] + Bc[t]. One wave owns a 16t x 256e strip; all
// M_prev values are preloaded into registers so the inner loop is pure
// WMMA + clause of 8 row stores (2x64B contiguous segments each).
// ---------------------------------------------------------------------------
__global__ __launch_bounds__(256) void outer_wmma_kernel(
    const float* __restrict__ Avals, const float* __restrict__ Bvals,
    const float* __restrict__ mprev, float* __restrict__ out) {
  const int wave = threadIdx.x >> 5;
  const int lane = threadIdx.x & 31;
  const int t0 = (blockIdx.x * 8 + wave) * 16;   // row tile start
  const int ebase = blockIdx.y * 256;            // this wave's e-strip
  const bool lo = lane < 16;
  const int l15 = lane & 15;
  const int rowoff = (lane >> 4) * 8;            // D rows: reg r -> M=r or r+8

  // Unconditional loads (valid for all lanes), branch-free mask for hi lanes.
  const float av = Avals[t0 + l15];
  const float bv = Bvals[t0 + l15];
  v2f afrag;
  afrag.x = lo ? av : 0.0f;  // K=0 column: A[t]
  afrag.y = lo ? bv : 0.0f;  // K=1 column: Bc[t]

  // Preload this lane's 16 M_prev values (hi lanes duplicate lo lanes; the
  // values feed zeroed A columns so they are don't-cares).
  float m[16];
#pragma unroll
  for (int i = 0; i < 16; ++i) {
    m[i] = mprev[ebase + i * 16 + l15];
  }

  float* const pbase = out + (size_t)(t0 + rowoff) * E_DIM + ebase + l15;
#pragma unroll
  for (int i = 0; i < 16; ++i) {
    v2f bfrag;
    bfrag.x = m[i];   // K=0 row: M_prev[e]
    bfrag.y = 1.0f;   // K=1 row: ones
    v8f c = {};
    // (neg_a, A, neg_b, B, c_mod, C, reuse_a, reuse_b)
    v8f d = __builtin_amdgcn_wmma_f32_16x16x4_f32(
        false, afrag, false, bfrag, (short)0, c, false, false);

    float* p = pbase + i * 16;
#pragma unroll
    for (int r = 0; r < 8; ++r) {
      p[(size_t)r * E_DIM] = d[r];
    }
  }
}

// ---------------------------------------------------------------------------
extern "C" void kernel_launch(void* const* d_in, const int* in_sizes, int n_in,
                              void* d_out, int out_size, void* d_ws,
                              size_t ws_size, hipStream_t stream) {
  const long long* indices = (const long long*)d_in[0];  // (T,B) int64
  const float* M_prev = (const float*)d_in[1];           // (E,) f32
  const float* params = (const float*)d_in[2];           // (N,2) f32
  float* out = (float*)d_out;                            // (T,E) f32

  float* ws = (float*)d_ws;        // needs 4*T floats = 256 KB
  float* alpha = ws;
  float* beta = ws + T_DIM;
  float* Asc = ws + 2 * T_DIM;
  float* Bsc = ws + 3 * T_DIM;

  coeff_reduce_kernel<<<T_DIM / 8, 256, 0, stream>>>(indices, params, alpha,
                                                     beta);
  affine_scan_kernel<<<1, 1024, 0, stream>>>(alpha, beta, Asc, Bsc);

  dim3 grid(T_DIM / 16 / 8, E_DIM / 256);  // (128, 16), 8 waves/block
  outer_wmma_kernel<<<grid, 256, 0, stream>>>(Asc, Bsc, M_prev, out);
}